// WeightedHausdorffDistance_70059506532413
// MI455X (gfx1250) — compile-verified
//
#include <hip/hip_runtime.h>
#include <math.h>

// Problem constants (match reference)
#define H_ 512
#define W_ 512
#define B_ 4
#define M_ 100
#define N_ (H_ * W_)
#define MT 7                       // ceil(100/16) m-tiles -> 112 padded columns
#define MCOLS (MT * 16)
#define MAX_DIST_F 724.07734394f   // sqrt(512^2 + 512^2)
#define EPS_F 1e-6f

#define WAVES_PER_BLOCK 8
#define PIX_TILES 16               // 16-pixel tiles per wave -> 256 pixels/wave
#define PIX_PER_WAVE (16 * PIX_TILES)
#define BLOCKS_PER_BATCH (N_ / (WAVES_PER_BLOCK * PIX_PER_WAVE))   // 128

typedef float v2f __attribute__((ext_vector_type(2)));
typedef float v8f __attribute__((ext_vector_type(8)));

// Workspace layout (floats): [0..B) sumP, [B..2B) sum(p*mind), [2B..2B+B*112) S[m]
#define WS_FLOATS (2 * B_ + B_ * MCOLS)

__global__ void whd_init(float* __restrict__ acc) {
    int tid = blockIdx.x * blockDim.x + threadIdx.x;
    if (tid < WS_FLOATS) acc[tid] = 0.0f;
}

__global__ __launch_bounds__(256) void whd_main(const float* __restrict__ prob,
                                                const float* __restrict__ gt,
                                                const float* __restrict__ osz,
                                                float* __restrict__ acc) {
    __shared__ float S_lds[MCOLS];
    __shared__ float t1_lds[WAVES_PER_BLOCK];
    __shared__ float p_lds[WAVES_PER_BLOCK];

    const int tid  = threadIdx.x;
    const int lane = tid & 31;
    const int wave = tid >> 5;
    const int half = lane >> 4;      // 0: lanes 0-15, 1: lanes 16-31
    const int l15  = lane & 15;

    const int b      = blockIdx.x / BLOCKS_PER_BATCH;
    const int blkInB = blockIdx.x % BLOCKS_PER_BATCH;

    if (tid < MCOLS) S_lds[tid] = 0.0f;
    __syncthreads();

    const float normY = osz[b * 2 + 0] * (1.0f / (float)H_);
    const float normX = osz[b * 2 + 1] * (1.0f / (float)W_);

    // ---- Build B fragments for all 7 m-tiles (B col m: [-2ym, -2xm, 1, ym^2+xm^2]) ----
    // B 4x16 f32 layout: vgpr v, lanes 0-15 -> K=v, lanes 16-31 -> K=v+2; column = lane&15
    v2f bf[MT];
    float colMask[MT];
    #pragma unroll
    for (int t = 0; t < MT; ++t) {
        int m = t * 16 + l15;
        int mc = (m < M_) ? m : (M_ - 1);
        float gy = gt[((size_t)b * M_ + mc) * 2 + 0] * normY;
        float gx = gt[((size_t)b * M_ + mc) * 2 + 1] * normX;
        // sentinel for padded columns: far away so it never wins min
        gy = (m < M_) ? gy : 60000.0f;
        gx = (m < M_) ? gx : 60000.0f;
        colMask[t] = (m < M_) ? 1.0f : 0.0f;
        bf[t][0] = half ? 1.0f                : (-2.0f * gy);   // K2 : K0
        bf[t][1] = half ? (gy * gy + gx * gx) : (-2.0f * gx);   // K3 : K1
    }

    float sAcc[MT];
    #pragma unroll
    for (int t = 0; t < MT; ++t) sAcc[t] = 0.0f;
    float t1acc = 0.0f, pacc = 0.0f;

    const int waveBase = (blkInB * WAVES_PER_BLOCK + wave) * PIX_PER_WAVE;
    const float* pb = prob + (size_t)b * N_;

    for (int it = 0; it < PIX_TILES; ++it) {
        const int base = waveBase + it * 16;

        // ---- A fragment: 16 pixels, row r = lane&15, [yn, xn, yn^2+xn^2, 1] ----
        // A 16x4 f32 layout: lanes 0-15 carry (K0,K1), lanes 16-31 carry (K2,K3)
        const int pix = base + l15;
        const float yn = (float)(pix >> 9) * normY;   // pix / 512
        const float xn = (float)(pix & 511) * normX;  // pix % 512
        v2f a;
        a[0] = half ? (yn * yn + xn * xn) : yn;
        a[1] = half ? 1.0f : xn;

        // p for the 8 rows this lane owns in the D tile (rows v + 8*half)
        const float4 p0 = *reinterpret_cast<const float4*>(pb + base + half * 8);
        const float4 p1 = *reinterpret_cast<const float4*>(pb + base + half * 8 + 4);
        float pv[8] = {p0.x, p0.y, p0.z, p0.w, p1.x, p1.y, p1.z, p1.w};

        float mind[8];
        #pragma unroll
        for (int v = 0; v < 8; ++v) mind[v] = 3.0e38f;

        #pragma unroll
        for (int t = 0; t < MT; ++t) {
            v8f c = {};
            // D = A x B + 0 : one 16x16 tile of squared distances in a single WMMA
            c = __builtin_amdgcn_wmma_f32_16x16x4_f32(
                    false, a, false, bf[t], (short)0, c, false, false);
            const float cm = colMask[t];
            #pragma unroll
            for (int v = 0; v < 8; ++v) {
                float d2 = c[v];
                // raw v_sqrt_f32: ~1 ulp, no fixup tail
                float d  = __builtin_amdgcn_sqrtf(d2 > 0.0f ? d2 : 0.0f);
                mind[v]  = fminf(mind[v], d);
                float p  = pv[v];
                float wd = MAX_DIST_F + p * (d - MAX_DIST_F) + EPS_F; // (1-p)*MD + p*d + eps
                float w2 = wd * wd, w4 = w2 * w2, w8 = w4 * w4;
                // raw v_rcp_f32 for wd^-9
                sAcc[t] += cm * __builtin_amdgcn_rcpf(w8 * wd);
            }
        }

        // per-row min across the 16 lanes of this half
        #pragma unroll
        for (int off = 8; off >= 1; off >>= 1) {
            #pragma unroll
            for (int v = 0; v < 8; ++v)
                mind[v] = fminf(mind[v], __shfl_xor(mind[v], off, 16));
        }
        if (l15 == 0) {
            #pragma unroll
            for (int v = 0; v < 8; ++v) { t1acc += pv[v] * mind[v]; pacc += pv[v]; }
        }
    }

    // ---- wave reduction of term1 partials (nonzero only on lanes 0 and 16) ----
    #pragma unroll
    for (int off = 16; off >= 1; off >>= 1) {
        t1acc += __shfl_xor(t1acc, off, 32);
        pacc  += __shfl_xor(pacc,  off, 32);
    }
    if (lane == 0) { t1_lds[wave] = t1acc; p_lds[wave] = pacc; }

    // ---- fold the two row-halves of sAcc, then LDS-atomic per column ----
    #pragma unroll
    for (int t = 0; t < MT; ++t) {
        float s = sAcc[t] + __shfl_xor(sAcc[t], 16, 32);
        if (half == 0) atomicAdd(&S_lds[t * 16 + l15], s);
    }
    __syncthreads();

    if (tid == 0) {
        float t1 = 0.0f, ps = 0.0f;
        for (int w = 0; w < WAVES_PER_BLOCK; ++w) { t1 += t1_lds[w]; ps += p_lds[w]; }
        atomicAdd(&acc[B_ + b], t1);
        atomicAdd(&acc[b], ps);
    }
    if (tid < MCOLS) atomicAdd(&acc[2 * B_ + b * MCOLS + tid], S_lds[tid]);
}

__global__ __launch_bounds__(512) void whd_final(const float* __restrict__ acc,
                                                 float* __restrict__ out) {
    __shared__ float minn[B_ * MCOLS];
    const int tid = threadIdx.x;
    if (tid < B_ * MCOLS) {
        int m = tid % MCOLS;
        float v = 0.0f;
        if (m < M_) {
            float mean = acc[2 * B_ + tid] * (1.0f / (float)N_);
            v = powf(mean, (float)(-1.0 / 9.0));   // (^1/P), P = -9
        }
        minn[tid] = v;
    }
    __syncthreads();
    if (tid == 0) {
        float t1s = 0.0f, t2s = 0.0f;
        for (int b = 0; b < B_; ++b) {
            t1s += acc[B_ + b] / (acc[b] + EPS_F);
            float s = 0.0f;
            for (int m = 0; m < MCOLS; ++m) s += minn[b * MCOLS + m];
            t2s += s * (1.0f / (float)M_);
        }
        out[0] = (t1s + t2s) * (1.0f / (float)B_);
    }
}

extern "C" void kernel_launch(void* const* d_in, const int* in_sizes, int n_in,
                              void* d_out, int out_size, void* d_ws, size_t ws_size,
                              hipStream_t stream) {
    const float* prob = (const float*)d_in[0];
    const float* gt   = (const float*)d_in[1];
    const float* osz  = (const float*)d_in[2];
    float* out = (float*)d_out;
    float* acc = (float*)d_ws;

    whd_init<<<1, 512, 0, stream>>>(acc);
    whd_main<<<B_ * BLOCKS_PER_BATCH, 256, 0, stream>>>(prob, gt, osz, acc);
    whd_final<<<1, 512, 0, stream>>>(acc, out);
}